// QuantumGate_6004364280071
// MI455X (gfx1250) — compile-verified
//
#include <hip/hip_runtime.h>
#include <hip/hip_bf16.h>
#include <math.h>

// ---------------------------------------------------------------------------
// 8-qubit variational circuit, B=65536 samples.
//   Layer l:  per-sample RY(x_i*pi/2) embedding, then a fixed complex
//   256x256 matrix M_l (trainable RY/RZ + CNOT ring).
// Strategy:
//   K1: build W_l = M_l^T via wave-per-basis-state register simulation,
//       stored pre-swizzled in the WMMA B-fragment layout so K2's B loads
//       are single aligned b64 per lane (256 B contiguous per wave).
//   K2: per 32-sample workgroup (two 16-row WMMA M-tiles): embedded state in
//       LDS -> complex GEMM with V_WMMA_F32_16X16X4_F32, B-fragments reused
//       across both M-tiles (8 wmma per 2 global b64) -> in-place layer-2
//       embedding -> GEMM -> <Z_i> via ds_add_f32.
// ---------------------------------------------------------------------------

typedef float v2f __attribute__((ext_vector_type(2)));
typedef float v8f __attribute__((ext_vector_type(8)));

#define NQ    8
#define NS    256          // 2^NQ amplitudes
#define NLAY  2
#define TM    32           // samples per workgroup (two WMMA M-tiles)
#define THREADS 256

// B-fragment swizzle: element (k, n) of the 256x256 plane lives at
//   ((k>>2)*16 + (n>>4))*64 + (((k>>1)&1)*16 + (n&15))*2 + (k&1)
// so that in the GEMM, lane L of the wave reading K-block k/4, N-tile n>>4
// loads its v2f fragment at ((k>>2)*16 + nt)*64 + L*2.
__device__ __forceinline__ int bswz(int k, int n) {
  const int fraglane = (((k >> 1) & 1) << 4) | (n & 15);
  return (((k >> 2) << 4) + (n >> 4)) * 64 + fraglane * 2 + (k & 1);
}

// ---------------------------------------------------------------------------
// Kernel 1: one wave per (layer, basis-state k). State: 8 complex amps/lane,
// global amp index s = (lane<<3)|r : bits 7..3 = lane bits, 2..0 local.
// Wire i <-> bit (7-i). Writes row k of W_l (= column k of M_l), swizzled.
// ---------------------------------------------------------------------------
__global__ __launch_bounds__(256) void build_layer_mats(
    const float* __restrict__ wts, float* __restrict__ W)
{
  const int t    = threadIdx.x;
  const int wave = blockIdx.x * (blockDim.x >> 5) + (t >> 5);
  const int lane = t & 31;
  const int layer = wave >> 8;      // 0..1
  const int k     = wave & 255;     // basis column (K index of W)

  float re[8], im[8];
#pragma unroll
  for (int r = 0; r < 8; ++r) {
    re[r] = (((lane << 3) | r) == k) ? 1.f : 0.f;
    im[r] = 0.f;
  }

  const float* wl = wts + layer * (NQ * 2);

  // trainable RY then RZ on each wire
#pragma unroll
  for (int i = 0; i < NQ; ++i) {
    const int b = 7 - i;
    const float hy = 0.5f * wl[i * 2 + 0];
    const float cy = cosf(hy), sy = sinf(hy);
    if (b < 3) {                       // local qubit: pair within lane
      const int m = 1 << b;
#pragma unroll
      for (int r = 0; r < 8; ++r) {
        if (!(r & m)) {
          const int r1 = r | m;
          float a0 = re[r], a1 = re[r1];
          re[r] = cy * a0 - sy * a1;  re[r1] = sy * a0 + cy * a1;
          a0 = im[r]; a1 = im[r1];
          im[r] = cy * a0 - sy * a1;  im[r1] = sy * a0 + cy * a1;
        }
      }
    } else {                           // lane qubit: wave32 butterfly
      const int lm = 1 << (b - 3);
      const bool hi = (lane & lm) != 0;
#pragma unroll
      for (int r = 0; r < 8; ++r) {
        const float ore = __shfl_xor(re[r], lm, 32);
        const float oim = __shfl_xor(im[r], lm, 32);
        re[r] = hi ? (sy * ore + cy * re[r]) : (cy * re[r] - sy * ore);
        im[r] = hi ? (sy * oim + cy * im[r]) : (cy * im[r] - sy * oim);
      }
    }
    // RZ: diag(e^{-i th/2}, e^{+i th/2})
    const float hz = 0.5f * wl[i * 2 + 1];
    const float cz = cosf(hz), sz = sinf(hz);
#pragma unroll
    for (int r = 0; r < 8; ++r) {
      const int bit = (b < 3) ? ((r >> b) & 1) : ((lane >> (b - 3)) & 1);
      const float tt = bit ? -sz : sz;
      const float nr = cz * re[r] + tt * im[r];
      const float ni = cz * im[r] - tt * re[r];
      re[r] = nr; im[r] = ni;
    }
  }

  // CNOT ring: (0,1)..(6,7),(7,0)
#pragma unroll
  for (int g = 0; g < NQ; ++g) {
    const int ci = (g < 7) ? g : 7;
    const int ti = (g < 7) ? g + 1 : 0;
    const int bc = 7 - ci, bt = 7 - ti;
    if (bt < 3) {                      // target local: register permute
      const int tm = 1 << bt;
      float tre[8], tim[8];
#pragma unroll
      for (int r = 0; r < 8; ++r) { tre[r] = re[r]; tim[r] = im[r]; }
#pragma unroll
      for (int r = 0; r < 8; ++r) {
        const int cb = (bc < 3) ? ((r >> bc) & 1) : ((lane >> (bc - 3)) & 1);
        re[r] = cb ? tre[r ^ tm] : tre[r];
        im[r] = cb ? tim[r ^ tm] : tim[r];
      }
    } else {                           // target on lane bit: shuffle-select
      const int lm = 1 << (bt - 3);
#pragma unroll
      for (int r = 0; r < 8; ++r) {
        const float ore = __shfl_xor(re[r], lm, 32);
        const float oim = __shfl_xor(im[r], lm, 32);
        const int cb = (bc < 3) ? ((r >> bc) & 1) : ((lane >> (bc - 3)) & 1);
        re[r] = cb ? ore : re[r];
        im[r] = cb ? oim : im[r];
      }
    }
  }

  // row k of W_l = M_l e_k, stored in B-fragment swizzled layout
  float* Wre = W + layer * 2 * (NS * NS);
  float* Wim = Wre + NS * NS;
#pragma unroll
  for (int r = 0; r < 8; ++r) {
    const int n = (lane << 3) | r;     // N index of W (output amplitude)
    const int idx = bswz(k, n);
    Wre[idx] = re[r];
    Wim[idx] = im[r];
  }
}

// ---------------------------------------------------------------------------
// f32 WMMA helper: D = A(16x4) * B(4x16) + C, A/B 2 VGPRs, C/D 8 VGPRs.
// ---------------------------------------------------------------------------
static __device__ __forceinline__ v8f wmma_f32(v2f a, v2f b, v8f c) {
  return __builtin_amdgcn_wmma_f32_16x16x4_f32(
      /*neg_a=*/false, a, /*neg_b=*/false, b,
      /*c_mod=*/(short)0, c, /*reuse_a=*/false, /*reuse_b=*/false);
}

// ---------------------------------------------------------------------------
// Kernel 2: one workgroup (8 waves) per 32 samples (2 M-tiles).
// LDS state S: 32 x 256 complex (re/im planes, 64 KB).
// Wave w owns N-tiles {2w, 2w+1} and computes BOTH M-tiles for them so each
// B-fragment (one aligned b64 per lane, 256 B per wave) feeds 4 WMMAs.
// Accumulators (64 VGPRs) persist in registers across the read/write barrier
// so a single LDS buffer can be updated in place.
// ---------------------------------------------------------------------------
__global__ __launch_bounds__(THREADS) void circuit_gemm(
    const float* __restrict__ x, const float* __restrict__ W,
    float* __restrict__ out)
{
  __shared__ float Sre[TM][NS];
  __shared__ float Sim[TM][NS];
  __shared__ float csC[TM][NQ];
  __shared__ float csS[TM][NQ];
  __shared__ float oacc[TM][NQ];

  const int t  = threadIdx.x;
  const int s0 = blockIdx.x * TM;

  // per-sample embedding cos/sin (angle = x * pi/2, half-angle = x * pi/4)
  {
    const int m = t >> 3, i = t & 7;          // 256 threads = 32*8 entries
    const float a = x[(s0 + m) * NQ + i] * 0.785398163397448f;
    csC[m][i] = cosf(a);
    csS[m][i] = sinf(a);
    oacc[m][i] = 0.f;
  }
  __syncthreads();

  // layer-1 embedded state is a product state: amp(s) = prod_i (c_i|s_i)
  for (int e = t; e < TM * NS; e += THREADS) {
    const int m = e >> 8, s = e & 255;
    float p = 1.f;
#pragma unroll
    for (int i = 0; i < NQ; ++i)
      p *= ((s >> (7 - i)) & 1) ? csS[m][i] : csC[m][i];
    Sre[m][s] = p;
    Sim[m][s] = 0.f;
  }
  __syncthreads();

  const int wave = t >> 5, lane = t & 31;
  const int half = lane >> 4, lq = lane & 15;

  v8f accR[2][2], accI[2][2];   // [ntile][mtile]

  for (int l = 0; l < NLAY; ++l) {
    const float* Wre = W + l * 2 * (NS * NS);
    const float* Wim = Wre + NS * NS;

#pragma unroll
    for (int tt = 0; tt < 2; ++tt)
#pragma unroll
      for (int mt = 0; mt < 2; ++mt) {
        accR[tt][mt] = (v8f){0.f, 0.f, 0.f, 0.f, 0.f, 0.f, 0.f, 0.f};
        accI[tt][mt] = (v8f){0.f, 0.f, 0.f, 0.f, 0.f, 0.f, 0.f, 0.f};
      }

    // --- complex GEMM: S_new(32xN-tile) = S(32x256) * W(256xN-tile) ---
    for (int k = 0; k < NS; k += 4) {
      const int kk = k + 2 * half;              // A layout: lanes16-31 -> K=2,3
      const v2f ar0 = *(const v2f*)&Sre[lq][kk];        // M-tile 0
      const v2f ai0 = *(const v2f*)&Sim[lq][kk];
      const v2f ar1 = *(const v2f*)&Sre[lq + 16][kk];   // M-tile 1
      const v2f ai1 = *(const v2f*)&Sim[lq + 16][kk];
      const v2f an0 = -ai0;                     // f32 WMMA has no A-negate mod
      const v2f an1 = -ai1;
#pragma unroll
      for (int tt = 0; tt < 2; ++tt) {
        const int nt = wave * 2 + tt;
        // swizzled B-fragment: one aligned b64 per lane per plane
        const int fidx = (((k >> 2) << 4) + nt) * 64 + lane * 2;
        const v2f br = *(const v2f*)&Wre[fidx];
        const v2f bi = *(const v2f*)&Wim[fidx];
        accR[tt][0] = wmma_f32(ar0, br, accR[tt][0]);   // Re += Ar*Br
        accR[tt][0] = wmma_f32(an0, bi, accR[tt][0]);   // Re -= Ai*Bi
        accI[tt][0] = wmma_f32(ar0, bi, accI[tt][0]);   // Im += Ar*Bi
        accI[tt][0] = wmma_f32(ai0, br, accI[tt][0]);   // Im += Ai*Br
        accR[tt][1] = wmma_f32(ar1, br, accR[tt][1]);
        accR[tt][1] = wmma_f32(an1, bi, accR[tt][1]);
        accI[tt][1] = wmma_f32(ar1, bi, accI[tt][1]);
        accI[tt][1] = wmma_f32(ai1, br, accI[tt][1]);
      }
    }
    __syncthreads();   // all waves done reading S

    if (l == 0) {
      // write back (C layout: VGPR j -> M = j + 8*half, N = lq)
#pragma unroll
      for (int tt = 0; tt < 2; ++tt) {
        const int n = (wave * 2 + tt) * 16 + lq;
#pragma unroll
        for (int mt = 0; mt < 2; ++mt)
#pragma unroll
          for (int j = 0; j < 8; ++j) {
            const int m = mt * 16 + j + 8 * half;
            Sre[m][n] = accR[tt][mt][j];
            Sim[m][n] = accI[tt][mt][j];
          }
      }
      __syncthreads();
      // layer-2 embedding: 8 in-place pair-mixing passes
      for (int i = 0; i < NQ; ++i) {
        const int b = 7 - i;
        for (int p = t; p < TM * (NS / 2); p += THREADS) {
          const int m = p >> 7, pr = p & 127;
          const int low = pr & ((1 << b) - 1);
          const int sA = ((pr >> b) << (b + 1)) | low;
          const int sB = sA | (1 << b);
          const float c = csC[m][i], sn = csS[m][i];
          float a0 = Sre[m][sA], a1 = Sre[m][sB];
          Sre[m][sA] = c * a0 - sn * a1;
          Sre[m][sB] = sn * a0 + c * a1;
          a0 = Sim[m][sA]; a1 = Sim[m][sB];
          Sim[m][sA] = c * a0 - sn * a1;
          Sim[m][sB] = sn * a0 + c * a1;
        }
        __syncthreads();
      }
    }
  }

  // <Z_w> from final accumulators: sum_n |amp|^2 * (1 - 2*bit_w(n))
#pragma unroll
  for (int tt = 0; tt < 2; ++tt) {
    const int n = (wave * 2 + tt) * 16 + lq;
#pragma unroll
    for (int mt = 0; mt < 2; ++mt)
#pragma unroll
      for (int j = 0; j < 8; ++j) {
        const int m = mt * 16 + j + 8 * half;
        const float pv = accR[tt][mt][j] * accR[tt][mt][j] +
                         accI[tt][mt][j] * accI[tt][mt][j];
#pragma unroll
        for (int w = 0; w < NQ; ++w) {
          atomicAdd(&oacc[m][w], ((n >> (7 - w)) & 1) ? -pv : pv);
        }
      }
  }
  __syncthreads();

  {
    const int m = t >> 3, i = t & 7;            // 256 threads = 32*8 outputs
    out[(s0 + m) * NQ + i] = oacc[m][i];
  }
}

// ---------------------------------------------------------------------------
extern "C" void kernel_launch(void* const* d_in, const int* in_sizes, int n_in,
                              void* d_out, int out_size, void* d_ws, size_t ws_size,
                              hipStream_t stream) {
  const float* x   = (const float*)d_in[0];   // (65536, 8) f32
  const float* wts = (const float*)d_in[1];   // (2, 8, 2)  f32
  float* out = (float*)d_out;                 // (65536, 8) f32
  float* W   = (float*)d_ws;                  // 2 layers * 2 planes * 256*256 f32 = 1 MB

  const int B = in_sizes[0] / NQ;

  // 512 waves (2 layers x 256 basis states) = 64 blocks of 8 waves
  build_layer_mats<<<(NLAY * NS) / (256 / 32), 256, 0, stream>>>(wts, W);
  // one workgroup per 32 samples
  circuit_gemm<<<B / TM, THREADS, 0, stream>>>(x, W, out);
}